// MarbleAutogradLayer_79542794322071
// MI455X (gfx1250) — compile-verified
//
#include <hip/hip_runtime.h>

// out[i] = x[i] * prod_{j=0..15} weights[paths[i*16 + j]]
//
// Bandwidth-bound: paths (256 MB, int32, single-use) dominates HBM traffic.
// Strategy:
//   - NT-hinted 128-bit loads for the paths stream (don't pollute L2)
//   - NT load for x, NT store for out (single-use streams)
//   - regular (RT) gathers for weights -> stays resident in 192 MB L2 (8 MB set)
//   - 4 independent 4-term product chains for ILP
//   - global_prefetch_b8 warming the paths stream ahead

typedef int  int4v   __attribute__((ext_vector_type(4)));

#define PATH_LEN 16
#define BLOCK    256
#define PF_DIST  4096   // elements ahead (= 16 blocks) to warm L2 for the stream

__global__ __launch_bounds__(BLOCK) void wander_path_product_kernel(
    const float* __restrict__ x,
    const int*   __restrict__ paths,
    const float* __restrict__ weights,
    float*       __restrict__ out,
    int n)
{
    int i = blockIdx.x * BLOCK + threadIdx.x;
    if (i >= n) return;

    // Warm the paths stream for blocks that launch later.
    long long ahead = (long long)i + PF_DIST;
    if (ahead < (long long)n) {
        __builtin_prefetch(paths + ahead * (long long)PATH_LEN, 0, 1); // global_prefetch_b8
    }

    // 16 int32 indices = 64 contiguous bytes per element -> 4x b128 NT loads.
    const int4v* p = (const int4v*)(paths + (long long)i * PATH_LEN);
    int4v a = __builtin_nontemporal_load(p + 0);
    int4v b = __builtin_nontemporal_load(p + 1);
    int4v c = __builtin_nontemporal_load(p + 2);
    int4v d = __builtin_nontemporal_load(p + 3);

    // Four independent product chains (ILP across the 16 L2 gathers).
    float pa = (weights[a.x] * weights[a.y]) * (weights[a.z] * weights[a.w]);
    float pb = (weights[b.x] * weights[b.y]) * (weights[b.z] * weights[b.w]);
    float pc = (weights[c.x] * weights[c.y]) * (weights[c.z] * weights[c.w]);
    float pd = (weights[d.x] * weights[d.y]) * (weights[d.z] * weights[d.w]);

    float prod = (pa * pb) * (pc * pd);

    float xv = __builtin_nontemporal_load(x + i);
    __builtin_nontemporal_store(xv * prod, out + i);
}

extern "C" void kernel_launch(void* const* d_in, const int* in_sizes, int n_in,
                              void* d_out, int out_size, void* d_ws, size_t ws_size,
                              hipStream_t stream) {
    const float* x       = (const float*)d_in[0];  // 2048*2048 f32
    const int*   paths   = (const int*)  d_in[1];  // n*16 int32 indices
    const float* weights = (const float*)d_in[2];  // 2^21 f32
    float*       out     = (float*)d_out;

    int n = in_sizes[0];                            // 4,194,304 elements
    int grid = (n + BLOCK - 1) / BLOCK;             // 16,384 blocks of 8 wave32 waves

    wander_path_product_kernel<<<grid, BLOCK, 0, stream>>>(x, paths, weights, out, n);
}